// MinLoss_69707319214519
// MI455X (gfx1250) — compile-verified
//
#include <hip/hip_runtime.h>
#include <hip/hip_bf16.h>
#include <math.h>

typedef __attribute__((ext_vector_type(2))) float v2f;
typedef __attribute__((ext_vector_type(8))) float v8f;

// Problem constants from the reference
constexpr int Bn = 16;    // batch
constexpr int Tn = 4096;  // time
constexpr int Sn = 3;     // sources
constexpr int Dn = 512;   // feature dim

constexpr int TCHUNK = 64;              // t-values per workgroup
constexpr int NCHUNK = Tn / TCHUNK;     // 64 chunks per batch
constexpr int NWAVES = 8;               // 256 threads / wave32
constexpr int TPW    = TCHUNK / NWAVES; // 8 t-values per wave
// partial layout per WG: [0..8]=cross[s*3+r], [9..11]=pn[s], [12..14]=gn[r]
constexpr int PARTW = 15;

__global__ __launch_bounds__(256)
void minloss_cross_partial(const float* __restrict__ pred,
                           const float* __restrict__ gt,
                           float* __restrict__ part) {
    __shared__ float lds[NWAVES][16];

    const int wg    = blockIdx.x;
    const int b     = wg / NCHUNK;
    const int chunk = wg % NCHUNK;
    const int lane  = threadIdx.x & 31;
    const int wave  = threadIdx.x >> 5;

    // WMMA f32 16x16x4 fragment addressing (wave32):
    //   A (16x4): lane L, VGPR v -> A[M=L%16, K = v + 2*(L>>4)]
    //   B (4x16): lane L, VGPR v -> B[K = v + 2*(L>>4), N=L%16]
    // => both fragments: row (L%16), two consecutive floats at d0 + 2*(L>>4)
    const int m    = lane & 15;
    const int half = lane >> 4;
    const int mrow = (m < Sn) ? m : 0;   // clamp padding lanes to a valid row

    v8f c0 = {0.f,0.f,0.f,0.f,0.f,0.f,0.f,0.f};
    v8f c1 = {0.f,0.f,0.f,0.f,0.f,0.f,0.f,0.f};
    float ssp = 0.f;  // partial sum of P^2 for row (L%16)
    float ssg = 0.f;  // partial sum of G^2 for row (L%16)

    const int t0 = chunk * TCHUNK + wave * TPW;
    for (int ti = 0; ti < TPW; ++ti) {
        const int t = t0 + ti;
        const size_t base = ((size_t)(b * Tn + t) * Sn + mrow) * Dn + 2 * half;
        const float* pb = pred + base;
        const float* gb = gt   + base;
        // K loop over d: 2 WMMAs per iteration (independent accumulators)
        #pragma unroll 4
        for (int kk = 0; kk < Dn / 4; kk += 2) {
            v2f a0 = *(const v2f*)(pb + kk * 4);
            v2f b0 = *(const v2f*)(gb + kk * 4);
            v2f a1 = *(const v2f*)(pb + kk * 4 + 4);
            v2f b1 = *(const v2f*)(gb + kk * 4 + 4);
            ssp += a0.x * a0.x + a0.y * a0.y + a1.x * a1.x + a1.y * a1.y;
            ssg += b0.x * b0.x + b0.y * b0.y + b1.x * b1.x + b1.y * b1.y;
            // D = A * B + C  (fp32, exact same accumulation dtype as reference)
            c0 = __builtin_amdgcn_wmma_f32_16x16x4_f32(
                    false, a0, false, b0, (short)0, c0, false, false);
            c1 = __builtin_amdgcn_wmma_f32_16x16x4_f32(
                    false, a1, false, b1, (short)0, c1, false, false);
        }
    }
    v8f c = c0 + c1;

    // Fold K-halves of the wave: lane L and L+16 hold disjoint d-ranges of row m.
    ssp += __shfl_xor(ssp, 16, 32);
    ssg += __shfl_xor(ssg, 16, 32);

    // C layout: VGPR j, lane L -> C[M = j + 8*(L>>4), N = L%16].
    // So lane r (r<3) holds cross[s][r] in c[s] for s = 0..2.
    if (lane < Sn) {
        lds[wave][0 + lane] = c[0];   // cross[0][r]
        lds[wave][3 + lane] = c[1];   // cross[1][r]
        lds[wave][6 + lane] = c[2];   // cross[2][r]
        lds[wave][9 + lane]  = ssp;   // pn[lane]
        lds[wave][12 + lane] = ssg;   // gn[lane]
    }
    __syncthreads();

    if (threadIdx.x < PARTW) {
        float acc = 0.f;
        #pragma unroll
        for (int w = 0; w < NWAVES; ++w) acc += lds[w][threadIdx.x];
        part[(size_t)wg * PARTW + threadIdx.x] = acc;
    }
}

__global__ __launch_bounds__(256)
void minloss_finalize(const float* __restrict__ part, float* __restrict__ out) {
    __shared__ float red[Bn][PARTW];
    __shared__ float Dm[Bn][Sn][Sn];

    const int tid = threadIdx.x;
    if (tid < Bn * PARTW) {
        const int b = tid / PARTW;
        const int j = tid % PARTW;
        float s = 0.f;
        for (int ch = 0; ch < NCHUNK; ++ch)
            s += part[(size_t)(b * NCHUNK + ch) * PARTW + j];
        red[b][j] = s;
    }
    __syncthreads();

    if (tid < Bn * Sn * Sn) {
        const int b = tid / 9;
        const int s = (tid % 9) / 3;
        const int r = tid % 3;
        const float cr = red[b][s * 3 + r];
        const float pn = red[b][9 + s];
        const float gn = red[b][12 + r];
        Dm[b][s][r] = sqrtf(fmaxf(pn + gn - 2.0f * cr, 0.0f));
    }
    __syncthreads();

    if (tid == 0) {
        const int perms[6][3] = {{0,1,2},{0,2,1},{1,0,2},{1,2,0},{2,0,1},{2,1,0}};
        float best = INFINITY;
        for (int p = 0; p < 6; ++p) {
            float acc = 0.f;
            for (int b = 0; b < Bn; ++b) {
                float su = Dm[b][0][perms[p][0]]
                         + Dm[b][1][perms[p][1]]
                         + Dm[b][2][perms[p][2]];
                acc += fabsf(su * (1.0f / 3.0f));
            }
            best = fminf(best, acc * (1.0f / (float)Bn));
        }
        out[0] = logf(best);
    }
}

extern "C" void kernel_launch(void* const* d_in, const int* in_sizes, int n_in,
                              void* d_out, int out_size, void* d_ws, size_t ws_size,
                              hipStream_t stream) {
    (void)in_sizes; (void)n_in; (void)out_size; (void)ws_size;
    const float* pred = (const float*)d_in[0];
    const float* gt   = (const float*)d_in[1];
    float* part = (float*)d_ws;   // Bn*NCHUNK*PARTW floats = 60 KB

    minloss_cross_partial<<<dim3(Bn * NCHUNK), dim3(256), 0, stream>>>(pred, gt, part);
    minloss_finalize<<<dim3(1), dim3(256), 0, stream>>>(part, (float*)d_out);
}